// ClsSegLoss_76931454206569
// MI455X (gfx1250) — compile-verified
//
#include <hip/hip_runtime.h>
#include <math.h>

typedef float v2f __attribute__((ext_vector_type(2)));
typedef float v4f __attribute__((ext_vector_type(4)));
typedef float v8f __attribute__((ext_vector_type(8)));

#define NSAMP   64
#define NELEM   (512 * 512)          // per-sample C*H*W (C==1)
#define GROUPS  16                   // blocks per sample (matches 4 WMMA x K=4 chain)
#define CHUNK   (NELEM / GROUPS)     // 16384 floats per block
#define THREADS 256
#define F4_PER_THREAD (CHUNK / 4 / THREADS)  // 16 float4 loads per thread per array

// ---------------------------------------------------------------------------
// Stage 1: streaming sigmoid + per-block partial sums (bandwidth-bound).
// grid = (GROUPS, NSAMP), block = 256 threads (8 wave32).
// part[(s*GROUPS+g)*4 + {0,1,2,3}] = { sum s*m, sum s*s, sum m, sum s }
// ---------------------------------------------------------------------------
__global__ __launch_bounds__(THREADS)
void ClsSegLoss_stage1_partials(const float* __restrict__ seg,
                                const float* __restrict__ msk,
                                float* __restrict__ part) {
    const int g   = blockIdx.x;
    const int s   = blockIdx.y;
    const int tid = threadIdx.x;
    const size_t base = (size_t)s * NELEM + (size_t)g * CHUNK;
    const v4f* __restrict__ ps = (const v4f*)(seg + base);
    const v4f* __restrict__ pm = (const v4f*)(msk + base);

    float pg = 0.0f, pp = 0.0f, gg = 0.0f, sm = 0.0f;

#pragma unroll
    for (int i = 0; i < F4_PER_THREAD; ++i) {
        v4f x = __builtin_nontemporal_load(ps + i * THREADS + tid);
        v4f m = __builtin_nontemporal_load(pm + i * THREADS + tid);
#pragma unroll
        for (int c = 0; c < 4; ++c) {
            float xe = x[c];
            float me = m[c];
            // sigmoid via v_exp_f32 + v_rcp_f32 (trans pipe co-executes with VALU)
            float sv = __builtin_amdgcn_rcpf(1.0f + __expf(-xe));
            pg = fmaf(sv, me, pg);   // s * m
            pp = fmaf(sv, sv, pp);   // s * s
            gg += me;                // m * m == m (binary mask)
            sm += sv;                // s
        }
    }

    // wave32 shfl-xor tree reduction (deterministic)
#pragma unroll
    for (int off = 16; off > 0; off >>= 1) {
        pg += __shfl_xor(pg, off, 32);
        pp += __shfl_xor(pp, off, 32);
        gg += __shfl_xor(gg, off, 32);
        sm += __shfl_xor(sm, off, 32);
    }

    __shared__ float wsum[8][4];
    const int wave = tid >> 5;
    const int lane = tid & 31;
    if (lane == 0) {
        wsum[wave][0] = pg; wsum[wave][1] = pp;
        wsum[wave][2] = gg; wsum[wave][3] = sm;
    }
    __syncthreads();
    if (tid == 0) {
        float a0 = 0.f, a1 = 0.f, a2 = 0.f, a3 = 0.f;
#pragma unroll
        for (int w = 0; w < 8; ++w) {
            a0 += wsum[w][0]; a1 += wsum[w][1];
            a2 += wsum[w][2]; a3 += wsum[w][3];
        }
        float* o = part + ((size_t)s * GROUPS + g) * 4;
        o[0] = a0; o[1] = a1; o[2] = a2; o[3] = a3;
    }
}

// ---------------------------------------------------------------------------
// Stage 2: one wave32. Fold the 16 group-partials per (sample, component)
// with chained V_WMMA_F32_16X16X4_F32 against B = ones(4x16):
//   D[m][n] = sum_k A[m][k] * 1  (chained 4x over K -> sum of 16, fp32 exact
//   accumulate path). Then compute BCE + dice selection and write 2 outputs.
// ---------------------------------------------------------------------------
__global__ __launch_bounds__(32)
void ClsSegLoss_stage2_final(const float* __restrict__ part,
                             const float* __restrict__ predict_cls,
                             const int*   __restrict__ labels,
                             float* __restrict__ out) {
    __shared__ float R[NSAMP * 4];

    const int lane = threadIdx.x;        // 0..31, EXEC all ones for WMMA
    const int m  = lane & 15;            // A-matrix row (M)
    const int kb = (lane >> 4) * 2;      // lanes 0-15: K=0,1 ; lanes 16-31: K=2,3

    v2f bones = {1.0f, 1.0f};            // B = ones(4x16), layout-independent

    for (int q = 0; q < 4; ++q) {        // sample quarter: s = 16*q + m
        for (int c = 0; c < 4; ++c) {    // component
            v8f acc = {};
#pragma unroll
            for (int j = 0; j < 4; ++j) {    // chain: groups g = 4*j + k
                const int srow = q * 16 + m;
                const int g0   = 4 * j + kb;
                v2f a;
                a.x = part[((size_t)srow * GROUPS + g0) * 4 + c];
                a.y = part[((size_t)srow * GROUPS + g0 + 1) * 4 + c];
                acc = __builtin_amdgcn_wmma_f32_16x16x4_f32(
                    /*neg_a=*/false, a, /*neg_b=*/false, bones,
                    /*c_mod=*/(short)0, acc, /*reuse_a=*/false, /*reuse_b=*/false);
            }
            // D column n=0 lives in lane 0 (M = r) and lane 16 (M = 8 + r)
            if (lane == 0) {
#pragma unroll
                for (int r = 0; r < 8; ++r) R[(q * 16 + r) * 4 + c] = acc[r];
            }
            if (lane == 16) {
#pragma unroll
                for (int r = 0; r < 8; ++r) R[(q * 16 + 8 + r) * 4 + c] = acc[r];
            }
        }
    }
    __syncthreads();

    float bce = 0.0f, nsel = 0.0f, dsum = 0.0f;
#pragma unroll
    for (int t = 0; t < 2; ++t) {
        const int sidx = lane + 32 * t;
        const float pg = R[sidx * 4 + 0];
        const float pp = R[sidx * 4 + 1];
        const float gg = R[sidx * 4 + 2];
        const float sm = R[sidx * 4 + 3];
        const float lab = (float)labels[sidx];

        const float dpos = (2.0f * pg + 1e-5f) / (pp + gg + 1e-5f);
        const float dneg = 25.0f / (sm + 25.0f);
        const float dice = (lab == 1.0f) ? dpos : dneg;

        const float pc  = predict_cls[sidx];
        const float sel = (pc >= 0.5f) ? 1.0f : 0.0f;
        nsel += sel;
        dsum += sel * dice;

        const float pcc = fminf(fmaxf(pc, 1e-7f), 1.0f - 1e-7f);
        bce += -(lab * logf(pcc) + (1.0f - lab) * logf(1.0f - pcc));
    }

#pragma unroll
    for (int off = 16; off > 0; off >>= 1) {
        bce  += __shfl_xor(bce,  off, 32);
        nsel += __shfl_xor(nsel, off, 32);
        dsum += __shfl_xor(dsum, off, 32);
    }

    if (lane == 0) {
        out[0] = bce / (float)NSAMP;                                   // cls_loss
        out[1] = (nsel > 0.0f) ? (nsel - dsum) / fmaxf(nsel, 1.0f)
                               : 1e-4f;                                // seg_loss
    }
}

// ---------------------------------------------------------------------------
extern "C" void kernel_launch(void* const* d_in, const int* in_sizes, int n_in,
                              void* d_out, int out_size, void* d_ws, size_t ws_size,
                              hipStream_t stream) {
    const float* predict_cls = (const float*)d_in[0];   // [64]
    const float* predict_seg = (const float*)d_in[1];   // [64,1,512,512]
    const int*   labels      = (const int*)  d_in[2];   // [64]
    const float* masks       = (const float*)d_in[3];   // [64,1,512,512]
    float*       out         = (float*)d_out;           // [2]: (cls_loss, seg_loss)
    float*       part        = (float*)d_ws;            // 64*16*4 floats = 16 KB

    dim3 grid(GROUPS, NSAMP);
    ClsSegLoss_stage1_partials<<<grid, THREADS, 0, stream>>>(predict_seg, masks, part);
    ClsSegLoss_stage2_final<<<1, 32, 0, stream>>>(part, predict_cls, labels, out);
}